// DiscretisedBNF_11974368822030
// MI455X (gfx1250) — compile-verified
//
#include <hip/hip_runtime.h>
#include <hip/hip_bf16.h>
#include <math.h>

// Problem constants (match reference)
#define B_ROWS 2048
#define D_DIM  4096
#define H_DIM  1024
#define TWO_D  8192

// Tiling
#define BM 128
#define BN 128
#define KT 16
#define LDA (KT + 4)   // 20 floats/row: 80B => 16B-aligned rows for b128 LDS writes,
                       // 8B-aligned v2f frag loads, decent bank spread
#define LDB (BN + 4)   // 132 floats/row: 528B => 16B-aligned rows

typedef __attribute__((ext_vector_type(2))) float v2f;
typedef __attribute__((ext_vector_type(4))) float v4f;
typedef __attribute__((ext_vector_type(8))) float v8f;
typedef __attribute__((ext_vector_type(4))) int   v4i;

typedef __attribute__((address_space(1))) v4i gv4i;  // global 16B chunk
typedef __attribute__((address_space(3))) v4i lv4i;  // LDS 16B chunk

// ---------------------------------------------------------------------------
// Async global->LDS copy (GLOBAL_LOAD_ASYNC_TO_LDS_B128, ASYNCcnt-tracked).
// Signature (from hipcc diagnostic): (v4i AS1* src, v4i AS3* dst, imm off, imm cpol)
// ---------------------------------------------------------------------------
#if __has_builtin(__builtin_amdgcn_global_load_async_to_lds_b128)
#define ASYNC_CP 1
#endif

__device__ __forceinline__ void cp16(void* l, const void* g) {
#ifdef ASYNC_CP
  __builtin_amdgcn_global_load_async_to_lds_b128((gv4i*)g, (lv4i*)l, 0, 0);
#else
  *(v4f*)l = *(const v4f*)g;
#endif
}

#ifdef ASYNC_CP
#if __has_builtin(__builtin_amdgcn_s_wait_asynccnt)
#define WAIT_ASYNC(n) __builtin_amdgcn_s_wait_asynccnt(n)
#else
#define WAIT_ASYNC(n) asm volatile("s_wait_asynccnt %0" ::"i"(n) : "memory")
#endif
#else
#define WAIT_ASYNC(n) ((void)0)
#endif

// ---------------------------------------------------------------------------
// WMMA f32 16x16x4 fragment conventions (CDNA5 ISA 7.12.2):
//  A (16x4): lane L<16: v0=A[L][0], v1=A[L][1]; lane L+16: v0=A[L][2], v1=A[L][3]
//  B (4x16): lane L<16: v0=B[0][L], v1=B[1][L]; lane L+16: v0=B[2][L], v1=B[3][L]
//  C (16x16): vgpr r: lanes0-15 -> C[r][lane]; lanes16-31 -> C[r+8][lane-16]
// ---------------------------------------------------------------------------

// Kernel 1: h = leakyrelu(mu @ W1[:D] + t*W1[D,:] + b1)   (2048x1024 fp32)
__global__ __launch_bounds__(256)
void bnf_gemm1_lrelu(const float* __restrict__ mu, const float* __restrict__ t,
                     const float* __restrict__ W1, const float* __restrict__ b1,
                     float* __restrict__ h_out) {
  __shared__ float As[2][BM * LDA];
  __shared__ float Bs[2][KT * LDB];

  const int tid = threadIdx.x;
  const int lane = tid & 31;
  const int laneLo = lane & 15;
  const int laneHi = lane >> 4;
  const int wave = tid >> 5;
  const int wm = wave & 1;   // 0..1  (64 rows each)
  const int wn = wave >> 1;  // 0..3  (32 cols each)

  const int m0 = blockIdx.x * BM;
  const int n0 = blockIdx.y * BN;

  // 4 async b128 per thread per tile (A: 2, B: 2)
  auto issue_tile = [&](int k0, int buf) {
#pragma unroll
    for (int i = 0; i < 2; ++i) {
      int idx = tid + i * 256;                 // 0..511
      int r = idx >> 2, c4 = (idx & 3) << 2;   // A: 128 rows x 16 cols
      cp16(&As[buf][r * LDA + c4], mu + (size_t)(m0 + r) * D_DIM + k0 + c4);
    }
#pragma unroll
    for (int i = 0; i < 2; ++i) {
      int idx = tid + i * 256;
      int r = idx >> 5, c4 = (idx & 31) << 2;  // B: 16 rows x 128 cols
      cp16(&Bs[buf][r * LDB + c4], W1 + (size_t)(k0 + r) * H_DIM + n0 + c4);
    }
  };

  v8f acc[4][2];
#pragma unroll
  for (int mi = 0; mi < 4; ++mi)
#pragma unroll
    for (int ni = 0; ni < 2; ++ni)
#pragma unroll
      for (int e = 0; e < 8; ++e) acc[mi][ni][e] = 0.0f;

  const int NT = D_DIM / KT;
  issue_tile(0, 0);
  for (int ti = 0; ti < NT; ++ti) {
    const int buf = ti & 1;
    if (ti + 1 < NT) {
      issue_tile((ti + 1) * KT, buf ^ 1);
      WAIT_ASYNC(4);   // current tile's 4 copies done; next tile's 4 in flight
    } else {
      WAIT_ASYNC(0);
    }
    __syncthreads();

    const float* Ab = As[buf];
    const float* Bb = Bs[buf];
#pragma unroll
    for (int kk = 0; kk < KT; kk += 4) {
      v2f af[4], bf[2];
#pragma unroll
      for (int mi = 0; mi < 4; ++mi) {
        int row = wm * 64 + mi * 16 + laneLo;
        af[mi] = *(const v2f*)(Ab + row * LDA + kk + laneHi * 2);
      }
#pragma unroll
      for (int ni = 0; ni < 2; ++ni) {
        int col = wn * 32 + ni * 16 + laneLo;
        int kr = kk + laneHi * 2;
        bf[ni].x = Bb[kr * LDB + col];
        bf[ni].y = Bb[(kr + 1) * LDB + col];
      }
#pragma unroll
      for (int mi = 0; mi < 4; ++mi)
#pragma unroll
        for (int ni = 0; ni < 2; ++ni)
          acc[mi][ni] = __builtin_amdgcn_wmma_f32_16x16x4_f32(
              false, af[mi], false, bf[ni], (short)0, acc[mi][ni], false, false);
    }
    __syncthreads();  // all waves done reading buf before it is overwritten
  }

  // Epilogue: fold 4097th input column (t) as rank-1 update, add b1, leaky ReLU.
#pragma unroll
  for (int mi = 0; mi < 4; ++mi)
#pragma unroll
    for (int ni = 0; ni < 2; ++ni) {
      int col = n0 + wn * 32 + ni * 16 + laneLo;
      float w1last = W1[(size_t)D_DIM * H_DIM + col];
      float bias = b1[col];
#pragma unroll
      for (int r = 0; r < 8; ++r) {
        int row = m0 + wm * 64 + mi * 16 + laneHi * 8 + r;
        float v = acc[mi][ni][r] + t[row] * w1last + bias;
        v = (v >= 0.0f) ? v : 0.01f * v;
        h_out[(size_t)row * H_DIM + col] = v;
      }
    }
}

// Kernel 2: dual GEMM (mu_eps and ln_sigma_eps halves of W2) + fused epilogue.
// Telescoped K-bin sum: out = 0.5*(1+erf((0.875 - mu_x)/(sigma_x*sqrt(2))))
__global__ __launch_bounds__(256)
void bnf_gemm2_epilogue(const float* __restrict__ h, const float* __restrict__ mu,
                        const float* __restrict__ gamma,
                        const float* __restrict__ W2, const float* __restrict__ b2,
                        float* __restrict__ out) {
  __shared__ float As[2][BM * LDA];
  __shared__ float B1s[2][KT * LDB];
  __shared__ float B2s[2][KT * LDB];

  const int tid = threadIdx.x;
  const int lane = tid & 31;
  const int laneLo = lane & 15;
  const int laneHi = lane >> 4;
  const int wave = tid >> 5;
  const int wm = wave & 1;
  const int wn = wave >> 1;

  const int m0 = blockIdx.x * BM;
  const int n0 = blockIdx.y * BN;

  // 6 async b128 per thread per tile (A: 2, B1: 2, B2: 2)
  auto issue_tile = [&](int k0, int buf) {
#pragma unroll
    for (int i = 0; i < 2; ++i) {
      int idx = tid + i * 256;
      int r = idx >> 2, c4 = (idx & 3) << 2;
      cp16(&As[buf][r * LDA + c4], h + (size_t)(m0 + r) * H_DIM + k0 + c4);
    }
#pragma unroll
    for (int i = 0; i < 2; ++i) {
      int idx = tid + i * 256;
      int r = idx >> 5, c4 = (idx & 31) << 2;
      const float* gsrc = W2 + (size_t)(k0 + r) * TWO_D + n0 + c4;
      cp16(&B1s[buf][r * LDB + c4], gsrc);
      cp16(&B2s[buf][r * LDB + c4], gsrc + D_DIM);
    }
  };

  v8f acc1[4][2], acc2[4][2];
#pragma unroll
  for (int mi = 0; mi < 4; ++mi)
#pragma unroll
    for (int ni = 0; ni < 2; ++ni)
#pragma unroll
      for (int e = 0; e < 8; ++e) { acc1[mi][ni][e] = 0.0f; acc2[mi][ni][e] = 0.0f; }

  const int NT = H_DIM / KT;
  issue_tile(0, 0);
  for (int ti = 0; ti < NT; ++ti) {
    const int buf = ti & 1;
    if (ti + 1 < NT) {
      issue_tile((ti + 1) * KT, buf ^ 1);
      WAIT_ASYNC(6);
    } else {
      WAIT_ASYNC(0);
    }
    __syncthreads();

    const float* Ab = As[buf];
    const float* B1b = B1s[buf];
    const float* B2b = B2s[buf];
#pragma unroll
    for (int kk = 0; kk < KT; kk += 4) {
      v2f af[4], bf1[2], bf2[2];
#pragma unroll
      for (int mi = 0; mi < 4; ++mi) {
        int row = wm * 64 + mi * 16 + laneLo;
        af[mi] = *(const v2f*)(Ab + row * LDA + kk + laneHi * 2);
      }
#pragma unroll
      for (int ni = 0; ni < 2; ++ni) {
        int col = wn * 32 + ni * 16 + laneLo;
        int kr = kk + laneHi * 2;
        bf1[ni].x = B1b[kr * LDB + col];
        bf1[ni].y = B1b[(kr + 1) * LDB + col];
        bf2[ni].x = B2b[kr * LDB + col];
        bf2[ni].y = B2b[(kr + 1) * LDB + col];
      }
#pragma unroll
      for (int mi = 0; mi < 4; ++mi)
#pragma unroll
        for (int ni = 0; ni < 2; ++ni) {
          acc1[mi][ni] = __builtin_amdgcn_wmma_f32_16x16x4_f32(
              false, af[mi], false, bf1[ni], (short)0, acc1[mi][ni], false, false);
          acc2[mi][ni] = __builtin_amdgcn_wmma_f32_16x16x4_f32(
              false, af[mi], false, bf2[ni], (short)0, acc2[mi][ni], false, false);
        }
    }
    __syncthreads();
  }

  const float inv_sqrt2 = 0.70710678118654752f;
#pragma unroll
  for (int mi = 0; mi < 4; ++mi)
#pragma unroll
    for (int ni = 0; ni < 2; ++ni) {
      int col = n0 + wn * 32 + ni * 16 + laneLo;
      float bias1 = b2[col];
      float bias2 = b2[D_DIM + col];
#pragma unroll
      for (int r = 0; r < 8; ++r) {
        int row = m0 + wm * 64 + mi * 16 + laneHi * 8 + r;
        float g = gamma[row];
        float s = sqrtf((1.0f - g) / g);
        float mu_eps = acc1[mi][ni][r] + bias1;
        float ln_se  = acc2[mi][ni][r] + bias2;
        float mu_x = mu[(size_t)row * D_DIM + col] / g - s * mu_eps;
        float sigma_x = s * expf(ln_se);
        float inv = inv_sqrt2 / sigma_x;
        // Telescoped sum over K bins: only G(0.875) survives (G(-1.125)=0).
        out[(size_t)row * D_DIM + col] = 0.5f * (1.0f + erff((0.875f - mu_x) * inv));
      }
    }
}

extern "C" void kernel_launch(void* const* d_in, const int* in_sizes, int n_in,
                              void* d_out, int out_size, void* d_ws, size_t ws_size,
                              hipStream_t stream) {
  const float* mu    = (const float*)d_in[0];
  const float* t     = (const float*)d_in[1];
  const float* gamma = (const float*)d_in[2];
  const float* W1    = (const float*)d_in[3];
  const float* b1    = (const float*)d_in[4];
  const float* W2    = (const float*)d_in[5];
  const float* b2    = (const float*)d_in[6];
  float* out = (float*)d_out;
  float* h   = (float*)d_ws;  // 2048*1024 fp32 = 8 MB scratch

  dim3 g1(B_ROWS / BM, H_DIM / BN);  // 16 x 8
  bnf_gemm1_lrelu<<<g1, 256, 0, stream>>>(mu, t, W1, b1, h);

  dim3 g2(B_ROWS / BM, D_DIM / BN);  // 16 x 32
  bnf_gemm2_epilogue<<<g2, 256, 0, stream>>>(h, mu, gamma, W2, b2, out);
}